// MultiScaleDynamicCensus_27522150432949
// MI455X (gfx1250) — compile-verified
//
#include <hip/hip_runtime.h>

// MultiScaleDynamicCensus — fully fused single kernel for gfx1250 (MI455X).
//
// Per 16x16 output tile (one 256-thread block = 8 wave32s):
//   Phase A: xr = relu(bn(w1*x + b1)) for the 20x20 halo region, computed with
//            V_WMMA_F32_16X16X4_F32 (pixels x C(=3,pad 4) x R(=8,pad 16)) and
//            written straight into LDS (xr never touches HBM).
//   Phase B: per-pixel learned census: 8x 3x3 + 24x 5x5 depthwise filters per
//            channel, sigmoid((neigh - center)*t), mean -> two output planes.
//
// Roofline: fused traffic ~12 MB (=0.5us @ 23.3 TB/s) vs ~6.3 GFLOP depthwise
// FMA + 151M sigmoids -> VALU-bound. Hence: xr lives in LDS, weights ride the
// scalar path (uniform s_load), sigmoid = 1 fma + v_exp_f32 + v_rcp_f32 + 2 add,
// and the GEMM-shaped 1x1 conv runs on the WMMA pipe. Tile-coordinate math in
// phase A uses wave-uniform decomposition + row-wrap select (no v_mul_hi divs).

typedef __attribute__((ext_vector_type(2))) float v2f;
typedef __attribute__((ext_vector_type(8))) float v8f;

#define IMG_H 384
#define IMG_W 384
#define NB    4
#define RCH   8        // hidden channels
#define TS    16       // output tile edge
#define HALO  2
#define TP    (TS + 2 * HALO)     // 20
#define TPIX  (TP * TP)           // 400 halo pixels
#define LPITCH 408                // padded LDS pitch (408 % 64 = 24 -> conflict-free ch stores)

#define LOG2E 1.44269504088896340736f

__global__ __launch_bounds__(256) void census_fused_kernel(
    const float* __restrict__ x,      // (4,3,384,384)
    const float* __restrict__ w1,     // (8,3,1,1)
    const float* __restrict__ b1,     // (8)
    const float* __restrict__ gma,    // (8)
    const float* __restrict__ bta,    // (8)
    const float* __restrict__ mu,     // (8)
    const float* __restrict__ var,    // (8)
    const float* __restrict__ w3,     // (64,1,3,3)
    const float* __restrict__ b3,     // (64)
    const float* __restrict__ w5,     // (192,1,5,5)
    const float* __restrict__ b5,     // (192)
    const float* __restrict__ t3p,    // scalar
    const float* __restrict__ t5p,    // scalar
    float* __restrict__ out)          // (4,2,384,384)
{
  __shared__ float lds[RCH][LPITCH]; // xr halo tile, channel-major, padded pitch

  const int tid  = threadIdx.x;
  const int lane = tid & 31;         // wave32
  const int wave = tid >> 5;         // 8 waves / block
  const int half = lane >> 4;        // lane group 0/1
  const int l16  = lane & 15;

  const int h0 = blockIdx.y * TS;
  const int w0 = blockIdx.x * TS;
  const int b  = blockIdx.z;

  const size_t HW = (size_t)IMG_H * IMG_W;
  const float* xb = x + (size_t)b * 3 * HW;

  // Hoist the (uniform) temperature loads so SMEM latency overlaps phase A.
  const float t3 = fmaxf(t3p[0], 0.1f);
  const float t5 = fmaxf(t5p[0], 0.1f);
  const float T3 = -LOG2E * t3;      // exp2 scaling folded into temperature
  const float T5 = -LOG2E * t5;

  // ---------------- Phase A: 1x1 conv + BN + ReLU via WMMA f32 16x16x4 ----------------
  // B matrix (K=4 x N=16): B[k][n] = w1[n*3+k] for k<3 && n<8, else 0.
  // Mirrors the ISA A layout: lanes 0-15 hold K0/K1 in vgpr0/1 (N=lane),
  // lanes 16-31 hold K2/K3 (N=lane-16).
  v2f bmat; bmat.x = 0.f; bmat.y = 0.f;
  {
    const int k0 = half * 2;
    if (l16 < RCH) {
      if (k0     < 3) bmat.x = w1[l16 * 3 + k0];
      if (k0 + 1 < 3) bmat.y = w1[l16 * 3 + k0 + 1];
    }
  }
  // Fold bias + BN into y = conv*An + Cn, channel n = l16.
  float An = 0.f, Cn = 0.f;
  if (l16 < RCH) {
    float sc = gma[l16] * __builtin_amdgcn_rsqf(var[l16] + 1e-5f);
    An = sc;
    Cn = (b1[l16] - mu[l16]) * sc + bta[l16];
  }

  // 400 halo pixels -> 25 WMMA tiles of 16 pixels, round-robin over 8 waves.
  // qbase/tx0/ty0 are wave-uniform (scalar math); per-pixel row index is a
  // single wrap-select since tx0+m <= 34 < 2*TP (no integer division on VALU).
  for (int t = wave; t < 25; t += 8) {
    const int qbase = t * 16;
    const int ty0 = qbase / TP;            // uniform -> s_mul_hi chain, once
    const int tx0 = qbase - ty0 * TP;

    // A matrix (M=16 pixels x K=4): lane row m=l16; vgpr0/1 = channels (half*2, half*2+1).
    v2f amat; amat.x = 0.f; amat.y = 0.f;
    {
      const int txm  = tx0 + l16;
      const int wrap = (txm >= TP) ? 1 : 0;
      const int gh = h0 + ty0 + wrap - HALO;
      const int gw = w0 + txm - wrap * TP - HALO;
      const bool inb = ((unsigned)gh < (unsigned)IMG_H) & ((unsigned)gw < (unsigned)IMG_W);
      if (inb) {
        const size_t off = (size_t)gh * IMG_W + gw;
        amat.x = xb[(size_t)(half * 2) * HW + off];        // c = 0 or 2
        if (half == 0) amat.y = xb[HW + off];              // c = 1 ; c=3 pad -> 0
      }
    }
    v8f acc = {};
    acc = __builtin_amdgcn_wmma_f32_16x16x4_f32(
        /*neg_a=*/false, amat, /*neg_b=*/false, bmat,
        /*c_mod=*/(short)0, acc, /*reuse_a=*/false, /*reuse_b=*/false);

    // D layout: vgpr j, lanes 0-15 -> pixel m=j, lanes 16-31 -> m=j+8; N = l16.
    if (l16 < RCH) {
#pragma unroll
      for (int j = 0; j < 8; ++j) {
        const int m    = j + half * 8;
        const int txm  = tx0 + m;
        const int wrap = (txm >= TP) ? 1 : 0;
        const int ghh = h0 + ty0 + wrap - HALO;
        const int gww = w0 + txm - wrap * TP - HALO;
        const bool ib = ((unsigned)ghh < (unsigned)IMG_H) & ((unsigned)gww < (unsigned)IMG_W);
        float y = fmaxf(acc[j] * An + Cn, 0.f);
        lds[l16][qbase + m] = ib ? y : 0.f;  // zero-pad xr itself outside the image
      }
    }
  }
  __syncthreads();

  // ---------------- Phase B: learned multi-scale census ----------------
  const int lx = tid & 15;          // output pixel within tile
  const int ly = tid >> 4;

  float acc3 = 0.f, acc5 = 0.f;

  for (int r = 0; r < RCH; ++r) {   // r is wave-uniform -> weight loads stay scalar
    // 5x5 neighborhood of channel r (center at nb[2][2]); rows 16 lanes apart
    // and half-waves 20 words apart -> conflict-free LDS reads.
    float nb[5][5];
#pragma unroll
    for (int i = 0; i < 5; ++i)
#pragma unroll
      for (int j = 0; j < 5; ++j)
        nb[i][j] = lds[r][(ly + i) * TP + (lx + j)];
    const float c = nb[2][2];

    // sigmoid((v-c)*t) = rcp(1 + exp2(v*T + O)), T = -t*log2e, O = -c*T
    const float O3 = -c * T3;
    const float O5 = -c * T5;

    const float* w3r = w3 + r * 8 * 9;
    const float* b3r = b3 + r * 8;
#pragma unroll
    for (int k = 0; k < 8; ++k) {
      float v = b3r[k];
#pragma unroll
      for (int i = 0; i < 3; ++i)
#pragma unroll
        for (int j = 0; j < 3; ++j)
          v = fmaf(w3r[k * 9 + i * 3 + j], nb[i + 1][j + 1], v);
      float e = __builtin_amdgcn_exp2f(fmaf(v, T3, O3));
      acc3 += __builtin_amdgcn_rcpf(1.0f + e);
    }

    const float* w5r = w5 + r * 24 * 25;
    const float* b5r = b5 + r * 24;
#pragma unroll
    for (int k = 0; k < 24; ++k) {
      float v = b5r[k];
#pragma unroll
      for (int i = 0; i < 5; ++i)
#pragma unroll
        for (int j = 0; j < 5; ++j)
          v = fmaf(w5r[k * 25 + i * 5 + j], nb[i][j], v);
      float e = __builtin_amdgcn_exp2f(fmaf(v, T5, O5));
      acc5 += __builtin_amdgcn_rcpf(1.0f + e);
    }
  }

  const int oh = h0 + ly, ow = w0 + lx;
  float* ob = out + (size_t)b * 2 * HW + (size_t)oh * IMG_W + ow;
  ob[0]  = acc3 * (1.0f / 64.0f);    // mean over 8 ch * 8 neighbors
  ob[HW] = acc5 * (1.0f / 192.0f);   // mean over 8 ch * 24 neighbors
}

extern "C" void kernel_launch(void* const* d_in, const int* in_sizes, int n_in,
                              void* d_out, int out_size, void* d_ws, size_t ws_size,
                              hipStream_t stream) {
  (void)in_sizes; (void)n_in; (void)d_ws; (void)ws_size; (void)out_size;
  const float* x   = (const float*)d_in[0];
  const float* w1  = (const float*)d_in[1];
  const float* b1  = (const float*)d_in[2];
  const float* gma = (const float*)d_in[3];
  const float* bta = (const float*)d_in[4];
  const float* mu  = (const float*)d_in[5];
  const float* var = (const float*)d_in[6];
  const float* w3  = (const float*)d_in[7];
  const float* b3  = (const float*)d_in[8];
  const float* w5  = (const float*)d_in[9];
  const float* b5  = (const float*)d_in[10];
  const float* t3  = (const float*)d_in[11];
  const float* t5  = (const float*)d_in[12];

  dim3 grid(IMG_W / TS, IMG_H / TS, NB);   // (24, 24, 4)
  census_fused_kernel<<<grid, 256, 0, stream>>>(
      x, w1, b1, gma, bta, mu, var, w3, b3, w5, b5, t3, t5, (float*)d_out);
}